// GATSample_43009802502555
// MI455X (gfx1250) — compile-verified
//
#include <hip/hip_runtime.h>
#include <math.h>

// ---------------------------------------------------------------------------
// GAT 2-layer pipeline for MI455X (gfx1250, wave32).
//   GEMMs      : V_WMMA_F32_16X16X4_F32, one wave -> 16x64 strip (4 accs)
//   Edge phase : CSR (dst-sorted) built once per call with int atomics;
//                aggregation = wave per node, register softmax, NO float
//                atomics (all hot arrays are L2-resident on the 192MB L2).
// ---------------------------------------------------------------------------

typedef __attribute__((ext_vector_type(2))) float v2f;
typedef __attribute__((ext_vector_type(8))) float v8f;

// C[M,64] = A[M,K] @ B[K,64], row-major. One wave computes a 16-row strip
// across all 64 columns (4 WMMA column tiles share the A registers).
__global__ void __launch_bounds__(256)
gat_gemm_wmma(const float* __restrict__ A, const float* __restrict__ B,
              float* __restrict__ C, int M, int K) {
  const int lane    = threadIdx.x & 31;
  const int wv      = threadIdx.x >> 5;
  const int strip   = blockIdx.x * 8 + wv;     // 16-row strip index
  const int nstrips = (M + 15) >> 4;
  if (strip >= nstrips) return;                // wave-uniform exit
  const int row0 = strip * 16;
  const int kh   = lane >> 4;                  // 0: K pair {k,k+1}, 1: {k+2,k+3}
  const int l15  = lane & 15;
  int mrow = row0 + l15;
  if (mrow >= M) mrow = M - 1;                 // clamp loads, EXEC stays full

  const float* arow = A + (size_t)mrow * K;
  v8f acc[4];
#pragma unroll
  for (int t = 0; t < 4; ++t) acc[t] = (v8f){0.f,0.f,0.f,0.f,0.f,0.f,0.f,0.f};

  for (int k = 0; k < K; k += 4) {
    // A 16x4 layout: lanes 0-15 hold K={k,k+1}, lanes 16-31 hold K={k+2,k+3}
    const v2f a = *(const v2f*)(arow + k + 2 * kh);
    const float* brow0 = B + (size_t)(k + 2 * kh) * 64 + l15;
#pragma unroll
    for (int t = 0; t < 4; ++t) {
      v2f b;
      b.x = brow0[t * 16];
      b.y = brow0[64 + t * 16];
      acc[t] = __builtin_amdgcn_wmma_f32_16x16x4_f32(
          false, a, false, b, (short)0, acc[t], false, false);
    }
  }

  // C/D layout: VGPR r -> M = row0 + 8*kh + r, N = t*16 + l15
  const int rbase = row0 + 8 * kh;
#pragma unroll
  for (int t = 0; t < 4; ++t) {
    float* cp = C + (size_t)rbase * 64 + t * 16 + l15;
#pragma unroll
    for (int r = 0; r < 8; ++r)
      if (rbase + r < M) cp[(size_t)r * 64] = acc[t][r];
  }
}

// One wave per node: el = z.al, er = z.ar over F=64 (2 features / lane).
__global__ void gat_node_attn(const float* __restrict__ z,
                              const float* __restrict__ al,
                              const float* __restrict__ ar,
                              float* __restrict__ el, float* __restrict__ er,
                              int N) {
  const int wid  = (int)((blockIdx.x * (size_t)blockDim.x + threadIdx.x) >> 5);
  const int lane = threadIdx.x & 31;
  if (wid >= N) return;
  const float* zr = z + (size_t)wid * 64;
  const float z0 = zr[lane], z1 = zr[lane + 32];
  float sl = z0 * al[lane] + z1 * al[lane + 32];
  float sr = z0 * ar[lane] + z1 * ar[lane + 32];
#pragma unroll
  for (int off = 16; off; off >>= 1) {
    sl += __shfl_xor(sl, off, 32);
    sr += __shfl_xor(sr, off, 32);
  }
  if (lane == 0) { el[wid] = sl; er[wid] = sr; }
}

// ----------------------------- CSR build -----------------------------------

__global__ void gat_zero_i32(int* __restrict__ p, int n) {
  const int i = blockIdx.x * blockDim.x + threadIdx.x;
  if (i < n) p[i] = 0;
}

__global__ void gat_histogram(const int* __restrict__ dst, int* __restrict__ deg, int E) {
  const int i = blockIdx.x * blockDim.x + threadIdx.x;
  if (i < E) atomicAdd(deg + dst[i], 1);
}

// Per-block sums of deg (256/block).
__global__ void gat_scan_partials(const int* __restrict__ deg, int* __restrict__ partial,
                                  int N) {
  __shared__ int sh[256];
  const int i = blockIdx.x * 256 + threadIdx.x;
  sh[threadIdx.x] = (i < N) ? deg[i] : 0;
  __syncthreads();
  for (int off = 128; off; off >>= 1) {
    if (threadIdx.x < off) sh[threadIdx.x] += sh[threadIdx.x + off];
    __syncthreads();
  }
  if (threadIdx.x == 0) partial[blockIdx.x] = sh[0];
}

// Sequential exclusive scan of the ~196 block partials (in place).
__global__ void gat_scan_offsets(int* __restrict__ partial, int nb) {
  if (blockIdx.x == 0 && threadIdx.x == 0) {
    int run = 0;
    for (int b = 0; b < nb; ++b) { const int t = partial[b]; partial[b] = run; run += t; }
  }
}

// Intra-block exclusive scan + block offset -> rowstart & cursor.
__global__ void gat_scan_block(const int* __restrict__ deg, const int* __restrict__ partial,
                               int* __restrict__ rowstart, int* __restrict__ cursor, int N) {
  __shared__ int sh[256];
  const int i = blockIdx.x * 256 + threadIdx.x;
  const int v = (i < N) ? deg[i] : 0;
  sh[threadIdx.x] = v;
  __syncthreads();
  int val = v;
  for (int off = 1; off < 256; off <<= 1) {
    const int t = (threadIdx.x >= off) ? sh[threadIdx.x - off] : 0;
    __syncthreads();
    val += t;
    sh[threadIdx.x] = val;
    __syncthreads();
  }
  if (i < N) {
    const int ex = partial[blockIdx.x] + val - v;   // exclusive prefix
    rowstart[i] = ex;
    cursor[i]   = ex;
  }
}

// Bucket edges by destination: ssrc[rowstart[d] .. ) = src of d's in-edges.
__global__ void gat_scatter(const int* __restrict__ src, const int* __restrict__ dst,
                            int* __restrict__ cursor, int* __restrict__ ssrc, int E) {
  const int i = blockIdx.x * blockDim.x + threadIdx.x;
  if (i < E) {
    const int pos = atomicAdd(cursor + dst[i], 1);
    ssrc[pos] = src[i];
  }
}

// ------------------------- fused softmax-aggregate -------------------------
// One wave per destination node. Lanes cover the 64 features (float2/lane).
// Attention logits are recomputed from el/er (4B gather) instead of being
// materialized; the 256B/edge z-row gather is the only heavy traffic and it
// streams with zero atomics. Bias (+ optional ReLU) fused into the store.
__global__ void gat_node_aggregate(const int* __restrict__ rowstart,
                                   const int* __restrict__ deg,
                                   const int* __restrict__ ssrc,
                                   const float* __restrict__ el,
                                   const float* __restrict__ er,
                                   const float* __restrict__ z,
                                   const float* __restrict__ bias,
                                   float* __restrict__ out,
                                   int N, int applyRelu) {
  const int wid  = (int)((blockIdx.x * (size_t)blockDim.x + threadIdx.x) >> 5);
  const int lane = threadIdx.x & 31;
  if (wid >= N) return;
  const int start = rowstart[wid];
  const int cnt   = deg[wid];
  const float ern = er[wid];

  // pass 1: segment max of leaky_relu(el[src]+er[n])
  float mx = -INFINITY;
  for (int j = lane; j < cnt; j += 32) {
    float v = el[ssrc[start + j]] + ern;
    v = (v > 0.f) ? v : 0.2f * v;
    mx = fmaxf(mx, v);
  }
#pragma unroll
  for (int off = 16; off; off >>= 1) mx = fmaxf(mx, __shfl_xor(mx, off, 32));

  // pass 2: softmax denominator
  float sum = 0.f;
  for (int j = lane; j < cnt; j += 32) {
    float v = el[ssrc[start + j]] + ern;
    v = (v > 0.f) ? v : 0.2f * v;
    sum += __expf(v - mx);
  }
#pragma unroll
  for (int off = 16; off; off >>= 1) sum += __shfl_xor(sum, off, 32);
  const float rcp = (cnt > 0) ? (1.f / sum) : 0.f;

  // pass 3: alpha-weighted sum of z[src] rows (wave reads one row / step)
  float2 acc = {0.f, 0.f};
  for (int j = 0; j < cnt; ++j) {
    const int sv = ssrc[start + j];
    float v = el[sv] + ern;                 // broadcast load (same addr/wave)
    v = (v > 0.f) ? v : 0.2f * v;
    const float alpha = __expf(v - mx) * rcp;
    const float2 zz = ((const float2*)(z + (size_t)sv * 64))[lane];
    acc.x += alpha * zz.x;
    acc.y += alpha * zz.y;
  }

  float2 o;
  o.x = acc.x + bias[2 * lane];
  o.y = acc.y + bias[2 * lane + 1];
  if (applyRelu) { o.x = fmaxf(o.x, 0.f); o.y = fmaxf(o.y, 0.f); }
  ((float2*)(out + (size_t)wid * 64))[lane] = o;
}

// ---------------------------------------------------------------------------

extern "C" void kernel_launch(void* const* d_in, const int* in_sizes, int n_in,
                              void* d_out, int out_size, void* d_ws, size_t ws_size,
                              hipStream_t stream) {
  const float* x   = (const float*)d_in[0];
  const float* W1  = (const float*)d_in[1];
  const float* al1 = (const float*)d_in[2];
  const float* ar1 = (const float*)d_in[3];
  const float* b1  = (const float*)d_in[4];
  const float* W2  = (const float*)d_in[5];
  const float* al2 = (const float*)d_in[6];
  const float* ar2 = (const float*)d_in[7];
  const float* b2  = (const float*)d_in[8];
  const int*   src = (const int*)d_in[9];
  const int*   dst = (const int*)d_in[10];
  float* out = (float*)d_out;

  const int F  = 64;                 // HID_F == OUT_F
  const int N  = out_size / F;       // 50000
  const int E  = in_sizes[9];        // 1200000
  const int K1 = in_sizes[0] / N;    // 128
  (void)n_in; (void)ws_size;

  // Workspace (all L2-resident): z 12.8MB, h 12.8MB, el/er 0.4MB,
  // CSR ints ~5.4MB -> ~31.5MB total.
  char* ws = (char*)d_ws;
  size_t off = 0;
  auto alloc = [&](size_t bytes) -> void* {
    void* p = ws + off;
    off += (bytes + 255) & ~(size_t)255;
    return p;
  };
  float* zbuf     = (float*)alloc((size_t)N * F * sizeof(float));
  float* hbuf     = (float*)alloc((size_t)N * F * sizeof(float));
  float* el       = (float*)alloc((size_t)N * sizeof(float));
  float* er       = (float*)alloc((size_t)N * sizeof(float));
  int*   deg      = (int*)alloc((size_t)N * sizeof(int));
  int*   rowstart = (int*)alloc((size_t)N * sizeof(int));
  int*   cursor   = (int*)alloc((size_t)N * sizeof(int));
  int*   ssrc     = (int*)alloc((size_t)E * sizeof(int));
  const int nb    = (N + 255) / 256;          // scan blocks (~196)
  int*   partial  = (int*)alloc((size_t)nb * sizeof(int));

  const int gN256  = (N + 255) / 256;
  const int gE256  = (E + 255) / 256;
  const int gWaveN = (N + 7) / 8;             // 256 thr = 8 waves / block
  const int gGemm  = (((N + 15) / 16) + 7) / 8;

  // ---- CSR build (once, shared by both layers) ----
  gat_zero_i32     <<<gN256, 256, 0, stream>>>(deg, N);
  gat_histogram    <<<gE256, 256, 0, stream>>>(dst, deg, E);
  gat_scan_partials<<<nb,    256, 0, stream>>>(deg, partial, N);
  gat_scan_offsets <<<1,      32, 0, stream>>>(partial, nb);
  gat_scan_block   <<<nb,    256, 0, stream>>>(deg, partial, rowstart, cursor, N);
  gat_scatter      <<<gE256, 256, 0, stream>>>(src, dst, cursor, ssrc, E);

  // ---- Layer 1: x[N,128] -> hbuf[N,64] (ReLU fused) ----
  gat_gemm_wmma     <<<gGemm,  256, 0, stream>>>(x, W1, zbuf, N, K1);
  gat_node_attn     <<<gWaveN, 256, 0, stream>>>(zbuf, al1, ar1, el, er, N);
  gat_node_aggregate<<<gWaveN, 256, 0, stream>>>(rowstart, deg, ssrc, el, er,
                                                 zbuf, b1, hbuf, N, /*relu=*/1);

  // ---- Layer 2: hbuf[N,64] -> out[N,64] ----
  gat_gemm_wmma     <<<gGemm,  256, 0, stream>>>(hbuf, W2, zbuf, N, F);
  gat_node_attn     <<<gWaveN, 256, 0, stream>>>(zbuf, al2, ar2, el, er, N);
  gat_node_aggregate<<<gWaveN, 256, 0, stream>>>(rowstart, deg, ssrc, el, er,
                                                 zbuf, b2, out, N, /*relu=*/0);
}